// NonLogicSparseGNN_77051713290427
// MI455X (gfx1250) — compile-verified
//
#include <hip/hip_runtime.h>

typedef __attribute__((ext_vector_type(2))) float v2f;
typedef __attribute__((ext_vector_type(8))) float v8f;

#define HF 64   // hidden width
#define FIN 40  // input features
#define NC 3    // classes

// ---------------- degree / normalization ----------------
__global__ void k_deg_init(float* __restrict__ deg, int n) {
  int i = blockIdx.x * blockDim.x + threadIdx.x;
  if (i < n) deg[i] = 1.0f;  // self-loop
}

__global__ void k_deg_edge(const int* __restrict__ edge, float* __restrict__ deg, int E) {
  int e = blockIdx.x * blockDim.x + threadIdx.x;
  if (e < E) atomicAdd(&deg[edge[E + e]], 1.0f);
}

__global__ void k_dinv(const float* __restrict__ deg, float* __restrict__ dinv, int n) {
  int i = blockIdx.x * blockDim.x + threadIdx.x;
  if (i < n) dinv[i] = rsqrtf(deg[i]);
}

// ---------------- GEMM1: h1 = x @ W1  (100000x40 @ 40x64), f32 WMMA ----------------
__global__ void __launch_bounds__(32) k_gemm1(const float* __restrict__ x,
                                              const float* __restrict__ W1,
                                              float* __restrict__ h1, int nTiles) {
  int tile = blockIdx.x;
  if (tile >= nTiles) return;             // uniform branch: EXEC stays all-ones
  int lane = threadIdx.x;
  int half = lane >> 4, lm = lane & 15;
  int r0 = tile * 16;
  int r = r0 + lm;                        // A row held by this lane

  // Preload A: 10 K-steps of 4; lanes 0-15 hold K=4k..4k+1, lanes 16-31 K=4k+2..4k+3
  v2f a[10];
#pragma unroll
  for (int k = 0; k < 10; ++k) {
    int kb = k * 4 + half * 2;
    a[k].x = x[r * FIN + kb];
    a[k].y = x[r * FIN + kb + 1];
  }

  v8f acc[4] = {};
#pragma unroll
  for (int nt = 0; nt < 4; ++nt) {
    int col = nt * 16 + lm;               // B column held by this lane
#pragma unroll
    for (int k = 0; k < 10; ++k) {
      int kb = k * 4 + half * 2;
      v2f b;
      b.x = W1[kb * HF + col];
      b.y = W1[(kb + 1) * HF + col];
      acc[nt] = __builtin_amdgcn_wmma_f32_16x16x4_f32(
          false, a[k], false, b, (short)0, acc[nt], false, false);
    }
  }

#pragma unroll
  for (int nt = 0; nt < 4; ++nt) {
    int col = nt * 16 + lm;
#pragma unroll
    for (int j = 0; j < 8; ++j) {         // C/D: VGPR j = row (half*8 + j)
      int row = r0 + half * 8 + j;
      h1[row * HF + col] = acc[nt][j];
    }
  }
}

// ---------------- init1: out1 = h1*dinv^2 + b1 (self-loop + bias) ----------------
__global__ void k_init1(const float* __restrict__ h1, const float* __restrict__ dinv,
                        const float* __restrict__ b1, float* __restrict__ out1, int total) {
  int t = blockIdx.x * blockDim.x + threadIdx.x;
  if (t < total) {
    int i = t >> 6, f = t & 63;
    float di = dinv[i];
    out1[t] = h1[t] * di * di + b1[f];
  }
}

// ---------------- agg1: out1[dst] += h1[src] * dinv[src]*dinv[dst] ----------------
__global__ void k_agg1(const int* __restrict__ edge, const float* __restrict__ h1,
                       const float* __restrict__ dinv, float* __restrict__ out1, int E) {
  long long t = (long long)blockIdx.x * blockDim.x + threadIdx.x;
  long long e = t >> 6;
  if (e >= E) return;
  int f = (int)(t & 63);
  int s = edge[e], d = edge[(long long)E + e];
  float nrm = dinv[s] * dinv[d];
  atomicAdd(&out1[(long long)d * HF + f], h1[(long long)s * HF + f] * nrm);
}

// ---------------- FC + spike + W2: f32 WMMA, then per-row dot via LDS ----------------
__global__ void __launch_bounds__(32) k_fc(const float* __restrict__ out1,
                                           const float* __restrict__ Wf,
                                           const float* __restrict__ bf,
                                           const float* __restrict__ W2,
                                           const float* __restrict__ ef,
                                           float* __restrict__ h2, int nTiles) {
  __shared__ float spk[16][HF];
  int tile = blockIdx.x;
  if (tile >= nTiles) return;
  int lane = threadIdx.x;
  int half = lane >> 4, lm = lane & 15;
  int r0 = tile * 16;
  int r = r0 + lm;

  // A = relu(out1 row), 16 K-steps of 4 (K = 64)
  v2f a[16];
#pragma unroll
  for (int k = 0; k < 16; ++k) {
    int kb = k * 4 + half * 2;
    float u = out1[r * HF + kb];
    float v = out1[r * HF + kb + 1];
    a[k].x = u > 0.0f ? u : 0.0f;
    a[k].y = v > 0.0f ? v : 0.0f;
  }

#pragma unroll
  for (int nt = 0; nt < 4; ++nt) {
    int col = nt * 16 + lm;
    v8f acc = {};
#pragma unroll
    for (int k = 0; k < 16; ++k) {
      int kb = k * 4 + half * 2;
      v2f b;
      b.x = Wf[kb * HF + col];
      b.y = Wf[(kb + 1) * HF + col];
      acc = __builtin_amdgcn_wmma_f32_16x16x4_f32(
          false, a[k], false, b, (short)0, acc, false, false);
    }
    float bc = bf[col];
#pragma unroll
    for (int j = 0; j < 8; ++j)
      spk[half * 8 + j][col] = (acc[j] + bc >= 0.5f) ? 1.0f : 0.0f;
  }

  __syncthreads();

  // h2[row] = (spike_row @ W2) * ef   (ethical filter folded; commutes with aggregation)
  if (lane < 16) {
#pragma unroll
    for (int c = 0; c < NC; ++c) {
      float s = 0.0f;
      for (int n = 0; n < HF; ++n) s += spk[lane][n] * W2[n * NC + c];
      h2[(r0 + lane) * NC + c] = s * ef[c];
    }
  }
}

// ---------------- init2: out = h2*dinv^2 + b2*ef ----------------
__global__ void k_init2(const float* __restrict__ h2, const float* __restrict__ dinv,
                        const float* __restrict__ b2, const float* __restrict__ ef,
                        float* __restrict__ out, int n) {
  int t = blockIdx.x * blockDim.x + threadIdx.x;
  if (t < n * NC) {
    int i = t / NC, c = t % NC;
    float di = dinv[i];
    out[t] = h2[t] * di * di + b2[c] * ef[c];
  }
}

// ---------------- agg2: out[dst] += h2[src] * norm (3 features/edge) ----------------
__global__ void k_agg2(const int* __restrict__ edge, const float* __restrict__ h2,
                       const float* __restrict__ dinv, float* __restrict__ out, int E) {
  int e = blockIdx.x * blockDim.x + threadIdx.x;
  if (e >= E) return;
  int s = edge[e], d = edge[(long long)E + e];
  float nrm = dinv[s] * dinv[d];
#pragma unroll
  for (int c = 0; c < NC; ++c)
    atomicAdd(&out[d * NC + c], h2[s * NC + c] * nrm);
}

extern "C" void kernel_launch(void* const* d_in, const int* in_sizes, int n_in,
                              void* d_out, int out_size, void* d_ws, size_t ws_size,
                              hipStream_t stream) {
  const float* x  = (const float*)d_in[0];
  const int*   ei = (const int*)d_in[1];   // JAX default x64-disabled -> int32
  const float* W1 = (const float*)d_in[2];
  const float* b1 = (const float*)d_in[3];
  const float* Wf = (const float*)d_in[4];
  const float* bf = (const float*)d_in[5];
  const float* W2 = (const float*)d_in[6];
  const float* b2 = (const float*)d_in[7];
  const float* ef = (const float*)d_in[8];
  float* out = (float*)d_out;

  const int N = in_sizes[0] / FIN;       // 100000
  const int E = in_sizes[1] / 2;         // 3200000
  const int nTiles = (N + 15) / 16;

  // Carve workspace (256B aligned): deg, dinv, h1, out1, h2  (~53 MB)
  char* base = (char*)d_ws;
  size_t off = 0;
  auto carve = [&](size_t bytes) {
    char* p = base + off;
    off += (bytes + 255) & ~(size_t)255;
    return p;
  };
  float* deg  = (float*)carve((size_t)N * 4);
  float* dinv = (float*)carve((size_t)N * 4);
  float* h1   = (float*)carve((size_t)N * HF * 4);
  float* out1 = (float*)carve((size_t)N * HF * 4);
  float* h2   = (float*)carve((size_t)N * NC * 4);
  (void)ws_size;

  const int B = 256;
  // 1) degrees + normalization
  k_deg_init<<<(N + B - 1) / B, B, 0, stream>>>(deg, N);
  k_deg_edge<<<(E + B - 1) / B, B, 0, stream>>>(ei, deg, E);
  k_dinv<<<(N + B - 1) / B, B, 0, stream>>>(deg, dinv, N);
  // 2) h1 = x @ W1 (WMMA f32)
  k_gemm1<<<nTiles, 32, 0, stream>>>(x, W1, h1, nTiles);
  // 3) conv1 aggregation
  k_init1<<<((N * HF) + B - 1) / B, B, 0, stream>>>(h1, dinv, b1, out1, N * HF);
  long long agg1T = (long long)E * HF;
  k_agg1<<<(unsigned)((agg1T + B - 1) / B), B, 0, stream>>>(ei, h1, dinv, out1, E);
  // 4) spike = (relu(out1) @ Wf + bf >= 0.5); h2 = spike @ W2 * ef (WMMA f32)
  k_fc<<<nTiles, 32, 0, stream>>>(out1, Wf, bf, W2, ef, h2, nTiles);
  // 5) conv2 aggregation straight into d_out
  k_init2<<<((N * NC) + B - 1) / B, B, 0, stream>>>(h2, dinv, b2, ef, out, N);
  k_agg2<<<(E + B - 1) / B, B, 0, stream>>>(ei, h2, dinv, out, E);
}